// Encoder_86646670230069
// MI455X (gfx1250) — compile-verified
//
#include <hip/hip_runtime.h>
#include <hip/hip_bf16.h>

typedef __attribute__((ext_vector_type(2))) float v2f;
typedef __attribute__((ext_vector_type(8))) float v8f;

// ---------------------------------------------------------------------------
// utility: grid-stride fill
// ---------------------------------------------------------------------------
__global__ void fill_f32(float* __restrict__ p, float v, size_t n) {
  size_t i = (size_t)blockIdx.x * blockDim.x + threadIdx.x;
  size_t stride = (size_t)gridDim.x * blockDim.x;
  for (; i < n; i += stride) p[i] = v;
}

// deg = 1 + scatter_add(1 @ dst)
__global__ void deg_accum(float* __restrict__ deg, const int* __restrict__ dst, int E) {
  int i = blockIdx.x * blockDim.x + threadIdx.x;
  if (i < E) atomicAdd(&deg[dst[i]], 1.0f);
}

__global__ void rsqrt_k(float* __restrict__ dis, const float* __restrict__ deg, int N) {
  int i = blockIdx.x * blockDim.x + threadIdx.x;
  if (i < N) dis[i] = rsqrtf(deg[i]);
}

// x[b,n,0:F_IN] = node; x[b,n,F_IN:128] = embed[node_type[n]]
__global__ void concat_k(float* __restrict__ x, const float* __restrict__ node,
                         const int* __restrict__ ntype, const float* __restrict__ embed,
                         int N, int F_IN, int EMB) {
  int n = blockIdx.x, b = blockIdx.y, f = threadIdx.x;
  float v;
  if (f < F_IN) v = node[((size_t)b * N + n) * F_IN + f];
  else          v = embed[ntype[n] * EMB + (f - F_IN)];
  x[(((size_t)b * N + n) << 7) + f] = v;
}

// ---------------------------------------------------------------------------
// C[M,128] = A[M,128] @ W[128,128], fp32 via V_WMMA_F32_16X16X4_F32.
// 256 threads = 8 waves; W staged in LDS; each wave: 16 rows x 128 cols.
// Fragment maps per CDNA5 ISA 7.12.2 (32-bit A 16x4; 32-bit C/D 16x16).
// ---------------------------------------------------------------------------
__global__ __launch_bounds__(256)
void gemm128_wmma(float* __restrict__ C, const float* __restrict__ A,
                  const float* __restrict__ W, int M) {
  __shared__ float Wlds[128 * 128];
  for (int i = threadIdx.x; i < (128 * 128) / 4; i += 256) {
    ((float4*)Wlds)[i] = ((const float4*)W)[i];
  }
  __syncthreads();

  const int wave = threadIdx.x >> 5;
  const int lane = threadIdx.x & 31;
  const int rowBase = (blockIdx.x * 8 + wave) * 16;
  if (rowBase >= M) return;                 // uniform per wave -> EXEC stays all-1s

  const int mrow   = rowBase + (lane & 15); // A-frag: lanes 0-15 K={0,1}, 16-31 K={2,3}
  const int khalf  = (lane >> 4) * 2;
  const int ncol   = lane & 15;             // B/C-frag column within tile

  v8f acc[8] = {};                          // 8 column tiles x 16x16 f32 accum

  const float* __restrict__ Arow = A + (size_t)mrow * 128;
  for (int k0 = 0; k0 < 128; k0 += 4) {
    v2f a;
    a.x = Arow[k0 + khalf + 0];
    a.y = Arow[k0 + khalf + 1];
#pragma unroll
    for (int t = 0; t < 8; ++t) {
      const int col = t * 16 + ncol;
      v2f b;
      b.x = Wlds[(k0 + khalf + 0) * 128 + col];
      b.y = Wlds[(k0 + khalf + 1) * 128 + col];
      acc[t] = __builtin_amdgcn_wmma_f32_16x16x4_f32(
          false, a, false, b, (short)0, acc[t], false, false);
    }
  }

  // C/D map: VGPR r -> row rowBase + r + (lane>=16 ? 8 : 0), col = tile*16 + (lane&15)
  const int rowHalf = (lane >> 4) * 8;
#pragma unroll
  for (int t = 0; t < 8; ++t) {
#pragma unroll
    for (int r = 0; r < 8; ++r) {
      C[(size_t)(rowBase + rowHalf + r) * 128 + t * 16 + ncol] = acc[t][r];
    }
  }
}

// ---------------------------------------------------------------------------
// agg[b,dst] += xw[b,src] * dis[src]*dis[dst]  (one wave per edge; lane = float4)
// 102 MB buffers fit in 192 MB L2 -> gathers and atomics resolve in L2.
// ---------------------------------------------------------------------------
__global__ __launch_bounds__(256)
void edge_agg(float* __restrict__ agg, const float* __restrict__ xw,
              const int* __restrict__ src, const int* __restrict__ dst,
              const float* __restrict__ dis, int E, int N, int B) {
  const int gw     = (blockIdx.x * blockDim.x + threadIdx.x) >> 5;
  const int lane   = threadIdx.x & 31;
  const int stride = (gridDim.x * blockDim.x) >> 5;
  for (int e = gw; e < E; e += stride) {
    const int s = src[e];
    const int d = dst[e];
    if (e + stride < E) {
      __builtin_prefetch(&xw[((size_t)src[e + stride]) << 7], 0, 0);
    }
    const float norm = dis[s] * dis[d];
#pragma unroll
    for (int b = 0; b < 2; ++b) {
      if (b >= B) break;
      const float4 v = reinterpret_cast<const float4*>(
          xw + (((size_t)b * N + s) << 7))[lane];
      float* a = agg + (((size_t)b * N + d) << 7) + lane * 4;
      atomicAdd(a + 0, v.x * norm);
      atomicAdd(a + 1, v.y * norm);
      atomicAdd(a + 2, v.z * norm);
      atomicAdd(a + 3, v.w * norm);
    }
  }
}

// h1 = relu(agg + xw * dis^2 + bias)   (self-loop term fused)
// grid (N, B) x block(128): indices come straight from IDs, no int division.
__global__ __launch_bounds__(128)
void finalize_relu(float* __restrict__ xout, const float* __restrict__ agg,
                   const float* __restrict__ xw, const float* __restrict__ dis,
                   const float* __restrict__ bias, int N) {
  const int n = blockIdx.x, b = blockIdx.y, h = threadIdx.x;
  const size_t idx = (((size_t)b * N + n) << 7) + h;
  const float dd = dis[n] * dis[n];
  const float v = agg[idx] + xw[idx] * dd + bias[h];
  xout[idx] = fmaxf(v, 0.0f);
}

// out[b,h] = mean_n(agg + xw*dis^2 + bias); block-partial sums, 1 atomic per chunk
__global__ __launch_bounds__(128)
void finalize_mean(float* __restrict__ out, const float* __restrict__ agg,
                   const float* __restrict__ xw, const float* __restrict__ dis,
                   const float* __restrict__ bias, int N, int chunk) {
  const int b = blockIdx.y;
  const int h = threadIdx.x;      // 0..127
  const int n0 = blockIdx.x * chunk;
  int n1 = n0 + chunk; if (n1 > N) n1 = N;
  const float bv = bias[h];
  float sum = 0.0f;
  for (int n = n0; n < n1; ++n) {
    const size_t idx = (((size_t)b * N + n) << 7) + h;
    const float dd = dis[n] * dis[n];
    sum += agg[idx] + xw[idx] * dd + bv;
  }
  atomicAdd(&out[b * 128 + h], sum / (float)N);
}

// ---------------------------------------------------------------------------
extern "C" void kernel_launch(void* const* d_in, const int* in_sizes, int n_in,
                              void* d_out, int out_size, void* d_ws, size_t ws_size,
                              hipStream_t stream) {
  const float* node  = (const float*)d_in[0];
  const int*   ntype = (const int*)d_in[1];
  const int*   eidx  = (const int*)d_in[2];
  const float* embed = (const float*)d_in[3];
  const float* W1    = (const float*)d_in[4];
  const float* b1    = (const float*)d_in[5];
  const float* W2    = (const float*)d_in[6];
  const float* b2    = (const float*)d_in[7];

  const int N    = in_sizes[1];
  const int E    = in_sizes[2] / 2;
  const int H    = in_sizes[5];          // 128
  const int EMB  = in_sizes[3] / 8;      // 12
  const int d1   = in_sizes[4] / H;      // 128 (= F_IN + EMB)
  const int F_IN = d1 - EMB;             // 116
  const int B    = in_sizes[0] / (N * F_IN);
  const int* src = eidx;
  const int* dst = eidx + E;
  const int  M   = B * N;
  const size_t BNH = (size_t)B * N * H;

  float* ws  = (float*)d_ws;
  size_t off = 0;
  float* deg = ws + off; off += (size_t)N;
  float* dis = ws + off; off += (size_t)N;
  float* X   = ws + off; off += BNH;
  float* XW  = ws + off; off += BNH;
  float* AGG = ws + off; off += BNH;
  (void)ws_size; (void)n_in;

  // --- degree / normalization -------------------------------------------
  fill_f32<<<1024, 256, 0, stream>>>(deg, 1.0f, (size_t)N);
  deg_accum<<<(E + 255) / 256, 256, 0, stream>>>(deg, dst, E);
  rsqrt_k<<<(N + 255) / 256, 256, 0, stream>>>(dis, deg, N);

  // --- layer 1 -----------------------------------------------------------
  concat_k<<<dim3(N, B), 128, 0, stream>>>(X, node, ntype, embed, N, F_IN, EMB);
  gemm128_wmma<<<(M + 127) / 128, 256, 0, stream>>>(XW, X, W1, M);
  fill_f32<<<4096, 256, 0, stream>>>(AGG, 0.0f, BNH);
  edge_agg<<<4096, 256, 0, stream>>>(AGG, XW, src, dst, dis, E, N, B);
  finalize_relu<<<dim3(N, B), 128, 0, stream>>>(X, AGG, XW, dis, b1, N);

  // --- layer 2 -----------------------------------------------------------
  gemm128_wmma<<<(M + 127) / 128, 256, 0, stream>>>(XW, X, W2, M);
  fill_f32<<<4096, 256, 0, stream>>>(AGG, 0.0f, BNH);
  edge_agg<<<4096, 256, 0, stream>>>(AGG, XW, src, dst, dis, E, N, B);

  // --- mean over nodes ---------------------------------------------------
  fill_f32<<<1, 256, 0, stream>>>((float*)d_out, 0.0f, (size_t)out_size);
  const int chunk   = 512;
  const int nchunks = (N + chunk - 1) / chunk;
  finalize_mean<<<dim3(nchunks, B), 128, 0, stream>>>((float*)d_out, AGG, XW, dis, b2, N, chunk);
}